// Toeplitz_11166914969892
// MI455X (gfx1250) — compile-verified
//
#include <hip/hip_runtime.h>
#include <hip/hip_bf16.h>
#include <stdint.h>

// out[b,s,e] = sum_{t<=s} weight[s-t] * x[b,t,e] + bias[s]
// Batched triangular-Toeplitz GEMM using V_WMMA_F32_16X16X4_F32 (exact fp32),
// GLOBAL_LOAD_ASYNC_TO_LDS staging (ASYNCcnt), row-pair-interleaved LDS layout
// (every WMMA B fragment = one aligned ds_load_b64), reversed weight vector
// (A fragment = one ascending ds_load_2addr_b32), scalar causal branch.

typedef float v2f __attribute__((ext_vector_type(2)));
typedef float v8f __attribute__((ext_vector_type(8)));

typedef __attribute__((address_space(1))) int g_int;   // global (device) int
typedef __attribute__((address_space(3))) int l_int;   // LDS (shared) int

#define T_DIM   2048
#define E_DIM   2048
#define KT      32          // K-rows staged per LDS chunk
#define NRP     (KT / 2)    // row pairs per chunk
#define RPSTR   288         // dwords per row pair: 128 cols * 2 + 32 pad
                            // (+32 dwords -> lane16-31 half hits disjoint banks)
#define WREV    2112        // reversed weight + 64 zero pad on top

#if defined(__has_builtin)
#if __has_builtin(__builtin_amdgcn_global_load_async_to_lds_b32)
#define HAVE_ASYNC_BUILTIN 1
#endif
#endif

__device__ static inline void async_b32(const float* g, const float* l)
{
#if defined(HAVE_ASYNC_BUILTIN)
    __builtin_amdgcn_global_load_async_to_lds_b32(
        (g_int*)(uintptr_t)g,
        (l_int*)(uint32_t)(uintptr_t)l,
        0, 0);
#else
    uint32_t lo = (uint32_t)(uintptr_t)l;
    asm volatile("global_load_async_to_lds_b32 %0, %1, off"
                 :: "v"(lo), "v"(g) : "memory");
#endif
}

__device__ static inline void wait_async0()
{
#if defined(__has_builtin)
#if __has_builtin(__builtin_amdgcn_s_wait_asynccnt)
    __builtin_amdgcn_s_wait_asynccnt(0);
#else
    asm volatile("s_wait_asynccnt 0x0" ::: "memory");
#endif
#else
    asm volatile("s_wait_asynccnt 0x0" ::: "memory");
#endif
}

__global__ __launch_bounds__(256) void
toeplitz_wmma_f32(const float* __restrict__ x,
                  const float* __restrict__ weight,
                  const float* __restrict__ bias,
                  float* __restrict__ out)
{
    __shared__ float Wr[WREV];              // Wr[j] = weight[2047-j], j<=2047; else 0
    __shared__ float Xs[2][NRP][RPSTR];     // interleaved: [rp][col*2 + (row&1)]

    const int tid  = threadIdx.x;
    const int lane = tid & 31;
    const int wave = tid >> 5;              // 8 waves -> 8 s-rows of 16
    const int eB   = blockIdx.x * 128;
    const int sB   = blockIdx.y * 128;
    const int b    = blockIdx.z;

    const float* xb = x + (size_t)b * T_DIM * E_DIM;

    // Staging decomposition: thread's chunk element (row = 2*it + r0, col)
    //   global offset = uniform(t0, it) + dvo   (dvo loop-invariant, divergent)
    //   lds   offset  = uniform(buf, it) + dlds (dlds loop-invariant, divergent)
    const int r0th = tid >> 7;              // 0 or 1
    const int colt = tid & 127;
    const int dvo  = r0th * E_DIM + colt;           // divergent global elem off
    const int dlds = (colt << 1) | r0th;            // divergent lds dword off

    // ---- issue async stage of chunk 0 : [KT x 128] floats ----
    {
        const float* cb = xb + eB;          // uniform base, t0 = 0
#pragma unroll
        for (int it = 0; it < 16; ++it)
            async_b32(cb + it * (2 * E_DIM) + dvo, &Xs[0][it][dlds]);
    }

    // ---- reversed, top-padded weight vector in LDS ----
    for (int i = tid; i < WREV; i += 256)
        Wr[i] = (i < T_DIM) ? weight[(T_DIM - 1) - i] : 0.0f;

    const int sWave = sB + wave * 16;
    const int sMaxS = __builtin_amdgcn_readfirstlane(sWave + 15); // scalar branch
    const int n  = lane & 15;               // N / M sub-lane index
    const int kh = lane >> 4;               // 0: K=0,1 half, 1: K=2,3 half
    // A fragment: idx(v0) = qr + t_global, idx(v1) = idx(v0)+1 (ascending!)
    const int qr = (T_DIM - 1) - sWave - n + kh * 2;

    v8f acc[8] = {};                        // 16 x 128 accumulator strip

    const int nchunks = sB / KT + (128 / KT);

    wait_async0();
    __syncthreads();                        // chunk 0 + Wr visible

    int p = 0;
    for (int c = 0; c < nchunks; ++c) {
        const int t0   = c * KT;
        const bool more = (c + 1 < nchunks);

        if (more) {                         // async-stage next chunk into Xs[p^1]
            const float* cb = xb + (size_t)(t0 + KT) * E_DIM + eB; // uniform
#pragma unroll
            for (int it = 0; it < 16; ++it)
                async_b32(cb + it * (2 * E_DIM) + dvo, &Xs[p ^ 1][it][dlds]);
        }

        // ---- compute: K-steps of 4 over this chunk (scalar causal skip) ----
#pragma unroll
        for (int r = 0; r < KT; r += 4) {
            const int tk = t0 + r;
            if (tk <= sMaxS) {
                v2f a;
                const int ja = qr + tk;
                a.x = Wr[ja];
                a.y = Wr[ja + 1];
                const float* bp = &Xs[p][(r >> 1) + kh][0];
#pragma unroll
                for (int j = 0; j < 8; ++j) {
                    const v2f bv = *(const v2f*)&bp[(j * 16 + n) << 1];
                    acc[j] = __builtin_amdgcn_wmma_f32_16x16x4_f32(
                        false, a, false, bv, (short)0, acc[j], false, false);
                }
            }
        }

        if (more) {
            wait_async0();                  // my async loads landed in LDS
            __syncthreads();                // everyone's loads landed; reads done
            p ^= 1;
        }
    }

    // ---- epilogue: bias add + store 16x128 strip ----
    float bl[8];
#pragma unroll
    for (int v = 0; v < 8; ++v)
        bl[v] = bias[sWave + v + kh * 8];

#pragma unroll
    for (int v = 0; v < 8; ++v) {
        const int srow2 = sWave + v + kh * 8;  // C layout: vgpr v -> row v / v+8
        float* orow = out + ((size_t)b * T_DIM + srow2) * E_DIM + eB;
#pragma unroll
        for (int j = 0; j < 8; ++j)
            orow[j * 16 + n] = acc[j][v] + bl[v];
    }
}

extern "C" void kernel_launch(void* const* d_in, const int* in_sizes, int n_in,
                              void* d_out, int out_size, void* d_ws, size_t ws_size,
                              hipStream_t stream)
{
    const float* x      = (const float*)d_in[0];
    const float* weight = (const float*)d_in[1];
    const float* bias   = (const float*)d_in[2];
    float*       out    = (float*)d_out;

    const int B = in_sizes[0] / (T_DIM * E_DIM);   // 8 for the reference shapes

    dim3 grid(E_DIM / 128, T_DIM / 128, B);        // (16, 16, 8)
    dim3 block(256);
    toeplitz_wmma_f32<<<grid, block, 0, stream>>>(x, weight, bias, out);
}